// ConvR_76381698392537
// MI455X (gfx1250) — compile-verified
//
#include <hip/hip_runtime.h>

typedef __attribute__((ext_vector_type(2))) float v2f;
typedef __attribute__((ext_vector_type(8))) float v8f;
typedef __attribute__((ext_vector_type(4))) int   v4i;

#define EMB 100
#define FS  5
#define RD  2500     // 100*5*5
#define NEN 100000
#define BATCH 256
#define FCIN 3600    // 100*6*6
#define BNEPS 1e-5f
#define NTILE 32     // entities per block (two 16-wide WMMA N-tiles)

// ---- async load-to-LDS (gfx1250 ASYNCcnt path), with safe fallback --------
#if defined(__has_builtin)
#  if __has_builtin(__builtin_amdgcn_global_load_async_to_lds_b128) && \
      __has_builtin(__builtin_amdgcn_s_wait_asynccnt)
#    define USE_ASYNC_LDS 1
#  endif
#endif
#ifndef USE_ASYNC_LDS
#  define USE_ASYNC_LDS 0
#endif

typedef __attribute__((address_space(1))) v4i gas_v4i;   // global int4
typedef __attribute__((address_space(3))) v4i las_v4i;   // LDS int4

// ---------------------------------------------------------------------------
// Kernel A: per-sample head pipeline (tiny: ~115 MFLOP total).
//   gather e1/rel -> BN -> dynamic-filter 5x5 conv (10x10 -> 100ch x 6x6)
//   -> BN1+relu -> FC(3600->100) -> BN2+relu -> h[b][e]
// ---------------------------------------------------------------------------
__global__ __launch_bounds__(128)
void convr_head(const float* __restrict__ emb_e, const float* __restrict__ emb_rel,
                const float* __restrict__ bnr_g, const float* __restrict__ bnr_b,
                const float* __restrict__ bnr_m, const float* __restrict__ bnr_v,
                const float* __restrict__ bn0_g, const float* __restrict__ bn0_b,
                const float* __restrict__ bn0_m, const float* __restrict__ bn0_v,
                const float* __restrict__ bn1_g, const float* __restrict__ bn1_b,
                const float* __restrict__ bn1_m, const float* __restrict__ bn1_v,
                const float* __restrict__ fc_w, const float* __restrict__ fc_b,
                const float* __restrict__ bn2_g, const float* __restrict__ bn2_b,
                const float* __restrict__ bn2_m, const float* __restrict__ bn2_v,
                const int* __restrict__ e1, const int* __restrict__ rel,
                float* __restrict__ hout)
{
    __shared__ float s_x0[EMB];     // BN0'd 10x10 input image
    __shared__ float s_f[RD];       // BN_rel'd filters, [c][k] k=i*5+j
    __shared__ float s_c[FCIN];     // conv output, index c*36 + h*6 + w

    const int b = blockIdx.x;
    const int t = threadIdx.x;
    const int ei = e1[b];
    const int ri = rel[b];

    const float s0 = bn0_g[0] * rsqrtf(bn0_v[0] + BNEPS);
    const float m0 = bn0_m[0], bb0 = bn0_b[0];
    for (int i = t; i < EMB; i += 128)
        s_x0[i] = (emb_e[(size_t)ei * EMB + i] - m0) * s0 + bb0;

    for (int i = t; i < RD; i += 128) {
        float r = emb_rel[(size_t)ri * RD + i];
        s_f[i] = (r - bnr_m[i]) * (bnr_g[i] * rsqrtf(bnr_v[i] + BNEPS)) + bnr_b[i];
    }
    __syncthreads();

    for (int o = t; o < FCIN; o += 128) {
        const int c  = o / 36;
        const int hw = o - c * 36;
        const int hh = hw / 6;
        const int ww = hw - hh * 6;
        const float* fc = &s_f[c * 25];
        float acc = 0.f;
        #pragma unroll
        for (int i = 0; i < FS; ++i)
            #pragma unroll
            for (int j = 0; j < FS; ++j)
                acc += s_x0[(hh + i) * 10 + (ww + j)] * fc[i * FS + j];
        acc = (acc - bn1_m[c]) * (bn1_g[c] * rsqrtf(bn1_v[c] + BNEPS)) + bn1_b[c];
        s_c[o] = acc > 0.f ? acc : 0.f;
    }
    __syncthreads();

    for (int e = t; e < EMB; e += 128) {
        float acc = fc_b[e];
        for (int i = 0; i < FCIN; ++i)
            acc += s_c[i] * fc_w[(size_t)i * EMB + e];
        acc = (acc - bn2_m[e]) * (bn2_g[e] * rsqrtf(bn2_v[e] + BNEPS)) + bn2_b[e];
        hout[(size_t)b * EMB + e] = acc > 0.f ? acc : 0.f;
    }
}

// ---------------------------------------------------------------------------
// Kernel B: scores = sigmoid(h(256x100) @ emb_e^T(100x100000) + bias)
// Store-bound (102.4 MB out). Block = 32-entity strip x all 256 batch rows.
// emb_e strip (32x100 f32 = 12.8 KB) staged in LDS ONCE per block via
// GLOBAL_LOAD_ASYNC_TO_LDS_B128 (ASYNCcnt), then all 8 waves serve B
// fragments from LDS. Each wave: 2 M-tiles x 2 N-tiles = 4 C tiles,
// 4 WMMA f32 16x16x4 per k-step, 25 k-steps.
// ---------------------------------------------------------------------------
__global__ __launch_bounds__(256)
void convr_scores(const float* __restrict__ h, const float* __restrict__ emb_e,
                  const float* __restrict__ bias, float* __restrict__ out)
{
    __shared__ float s_b[NTILE * EMB];         // 32 rows x 100 floats = 12.8 KB

    const int t    = threadIdx.x;
    const int wave = t >> 5;                   // 0..7
    const int lane = t & 31;
    const int l16  = lane & 15;
    const int khalf = (lane >> 4) << 1;        // 0 or 2
    const int n0 = blockIdx.x * NTILE;         // 3125 blocks
    const int m0 = wave * 32;

    // ---- stage emb_e[n0 .. n0+31][:] into LDS ----
#if USE_ASYNC_LDS
    {
        gas_v4i* g = (gas_v4i*)(void*)(emb_e + (size_t)n0 * EMB);
        las_v4i* l = (las_v4i*)(void*)s_b;
        // 32*100*4 = 12800 bytes = 800 x 16B chunks, 256 threads
        for (int i = t; i < (NTILE * EMB * 4) / 16; i += 256) {
            __builtin_amdgcn_global_load_async_to_lds_b128(g + i, l + i, 0, 0);
        }
        __builtin_amdgcn_s_wait_asynccnt(0);
    }
#else
    for (int i = t; i < NTILE * EMB; i += 256)
        s_b[i] = emb_e[(size_t)n0 * EMB + i];
#endif
    __syncthreads();

    const float* arow0 = h + (size_t)(m0 + l16) * EMB;
    const float* arow1 = h + (size_t)(m0 + 16 + l16) * EMB;
    const float* brow0 = s_b + l16 * EMB;
    const float* brow1 = s_b + (16 + l16) * EMB;

    v8f c00 = {}; v8f c01 = {}; v8f c10 = {}; v8f c11 = {};
    for (int k = 0; k < EMB; k += 4) {
        const int kk = k + khalf;
        v2f b0 = { brow0[kk], brow0[kk + 1] };
        v2f b1 = { brow1[kk], brow1[kk + 1] };
        v2f a0 = { arow0[kk], arow0[kk + 1] };
        v2f a1 = { arow1[kk], arow1[kk + 1] };
        c00 = __builtin_amdgcn_wmma_f32_16x16x4_f32(false, a0, false, b0,
                                                    (short)0, c00, false, false);
        c01 = __builtin_amdgcn_wmma_f32_16x16x4_f32(false, a0, false, b1,
                                                    (short)0, c01, false, false);
        c10 = __builtin_amdgcn_wmma_f32_16x16x4_f32(false, a1, false, b0,
                                                    (short)0, c10, false, false);
        c11 = __builtin_amdgcn_wmma_f32_16x16x4_f32(false, a1, false, b1,
                                                    (short)0, c11, false, false);
    }

    const int na = n0 + l16;
    const int nb = n0 + 16 + l16;
    const float ba = bias[na];
    const float bb = bias[nb];
    const int mrow = (lane < 16) ? 0 : 8;
    #pragma unroll
    for (int i = 0; i < 8; ++i) {
        const int m = m0 + mrow + i;
        float v;
        v = c00[i] + ba; out[(size_t)m * NEN + na]        = 1.f / (1.f + __expf(-v));
        v = c01[i] + bb; out[(size_t)m * NEN + nb]        = 1.f / (1.f + __expf(-v));
        v = c10[i] + ba; out[(size_t)(m + 16) * NEN + na] = 1.f / (1.f + __expf(-v));
        v = c11[i] + bb; out[(size_t)(m + 16) * NEN + nb] = 1.f / (1.f + __expf(-v));
    }
}

extern "C" void kernel_launch(void* const* d_in, const int* in_sizes, int n_in,
                              void* d_out, int out_size, void* d_ws, size_t ws_size,
                              hipStream_t stream) {
    const float* emb_e   = (const float*)d_in[0];
    const float* emb_rel = (const float*)d_in[1];
    const float* bnr_g   = (const float*)d_in[2];
    const float* bnr_b   = (const float*)d_in[3];
    const float* bnr_m   = (const float*)d_in[4];
    const float* bnr_v   = (const float*)d_in[5];
    const float* bn0_g   = (const float*)d_in[6];
    const float* bn0_b   = (const float*)d_in[7];
    const float* bn0_m   = (const float*)d_in[8];
    const float* bn0_v   = (const float*)d_in[9];
    const float* bn1_g   = (const float*)d_in[10];
    const float* bn1_b   = (const float*)d_in[11];
    const float* bn1_m   = (const float*)d_in[12];
    const float* bn1_v   = (const float*)d_in[13];
    const float* fc_w    = (const float*)d_in[14];
    const float* fc_b    = (const float*)d_in[15];
    const float* bn2_g   = (const float*)d_in[16];
    const float* bn2_b   = (const float*)d_in[17];
    const float* bn2_m   = (const float*)d_in[18];
    const float* bn2_v   = (const float*)d_in[19];
    const float* bias    = (const float*)d_in[20];
    const int*   e1      = (const int*)d_in[21];
    const int*   rel     = (const int*)d_in[22];

    float* hbuf = (float*)d_ws;              // 256*100 floats = 100 KB scratch
    float* outp = (float*)d_out;             // 256*100000 floats

    convr_head<<<BATCH, 128, 0, stream>>>(
        emb_e, emb_rel, bnr_g, bnr_b, bnr_m, bnr_v,
        bn0_g, bn0_b, bn0_m, bn0_v, bn1_g, bn1_b, bn1_m, bn1_v,
        fc_w, fc_b, bn2_g, bn2_b, bn2_m, bn2_v, e1, rel, hbuf);

    convr_scores<<<NEN / NTILE, 256, 0, stream>>>(hbuf, emb_e, bias, outp);
}